// ChaosKernel_71296457114141
// MI455X (gfx1250) — compile-verified
//
#include <hip/hip_runtime.h>
#include <hip/hip_bf16.h>
#include <math.h>

// ---------------------------------------------------------------------------
// MI455X (gfx1250) implementation of the recursive QFI-attention reference.
// B=1, T=768, D=64, L=4 layers, 2 passes.
//
// Heavy math (S·S^T pairwise inner, w·x attention apply) runs on
// V_WMMA_F32_16X16X4_F32. Wave32 fragment layouts per CDNA5 ISA 7.12.2.
// ---------------------------------------------------------------------------

typedef __attribute__((ext_vector_type(2))) float v2f;
typedef __attribute__((ext_vector_type(8))) float v8f;

constexpr int kT = 768;
constexpr int kD = 64;
constexpr int kL = 4;
constexpr int kNT = kT / 16;          // 48 column/row tiles
constexpr float kEPS = 1e-8f;

__device__ __forceinline__ float sigmoidf_(float v) {
    return 1.0f / (1.0f + __expf(-v));
}

// ---------------------------------------------------------------------------
// project_to_simplex + renorm + sqrt  ->  S = sqrt(p)   (one block per row)
// ---------------------------------------------------------------------------
__global__ void simplex_sqrt_kernel(const float* __restrict__ x, float* __restrict__ S) {
    __shared__ float red[kD];
    const int t = blockIdx.x;
    const int d = threadIdx.x;
    float v = x[t * kD + d];
    float sp = (v > 20.0f) ? v : log1pf(__expf(v));     // softplus
    red[d] = sp;
    __syncthreads();
    float s1 = 0.0f;
    #pragma unroll 8
    for (int i = 0; i < kD; ++i) s1 += red[i];
    float p = sp / (s1 + kEPS);
    p = fmaxf(p, kEPS);
    __syncthreads();
    red[d] = p;
    __syncthreads();
    float s2 = 0.0f;
    #pragma unroll 8
    for (int i = 0; i < kD; ++i) s2 += red[i];
    p = p / (s2 + kEPS);
    S[t * kD + d] = sqrtf(p);
}

// ---------------------------------------------------------------------------
// temperature: invT = 1 / max(sigmoid(basin . tw + tb) + 0.5, 1e-6)
// ---------------------------------------------------------------------------
__global__ void temp_kernel(const float* __restrict__ basin, const float* __restrict__ tw,
                            const float* __restrict__ tb, float* __restrict__ tempv) {
    __shared__ float red[kD];
    const int d = threadIdx.x;
    red[d] = basin[d] * tw[d];
    __syncthreads();
    if (d == 0) {
        float s = 0.0f;
        for (int i = 0; i < kD; ++i) s += red[i];
        float tm = sigmoidf_(s + tb[0]) + 0.5f;
        tempv[0] = 1.0f / fmaxf(tm, 1e-6f);
    }
}

// ---------------------------------------------------------------------------
// Feedback gate (passes > 0):  g = sigmoid([x,prev] @ W^T + b)
//                              x = x*g + prev*(1-g)
// One block per row, one thread per output dim.
// ---------------------------------------------------------------------------
__global__ void fb_gate_kernel(const float* __restrict__ x, const float* __restrict__ prev,
                               const float* __restrict__ W, const float* __restrict__ b,
                               float* __restrict__ xout) {
    const int t = blockIdx.x;
    const int d = threadIdx.x;
    const float* xr = x + t * kD;
    const float* pr = prev + t * kD;
    const float* wr = W + d * (2 * kD);
    float acc = b[d];
    #pragma unroll 8
    for (int k = 0; k < kD; ++k)
        acc += xr[k] * wr[k] + pr[k] * wr[kD + k];
    float g = sigmoidf_(acc);
    xout[t * kD + d] = xr[d] * g + pr[d] * (1.0f - g);
}

// ---------------------------------------------------------------------------
// Fused QFI attention for one 16-row tile per wave.
//   inner = S_tile . S^T          (WMMA f32 16x16x4, K=64)
//   logit = -2*acos(clip(inner)) * invT   -> scratch (L2)
//   softmax over 768 (two-pass, shfl reductions on wave32 halves)
//   O     = P . X                 (WMMA f32 16x16x4, K=768)
//   xout  = x + rs*(O/rowsum - x)
// Grid: 24 blocks x 64 threads (2 waves/block, 48 row tiles).
// ---------------------------------------------------------------------------
__global__ __launch_bounds__(64)
void attn_kernel(const float* __restrict__ x, const float* __restrict__ S,
                 float* __restrict__ logits, const float* __restrict__ tempv,
                 const float* __restrict__ rsl, int layer, float* __restrict__ xout) {
    __shared__ float tileBuf[16 * kD];      // staged 16x64 S or X tile (4 KB)
    __shared__ float Pt[2][16 * 17];        // per-wave padded P tile

    const int tid  = threadIdx.x;
    const int wave = tid >> 5;
    const int lane = tid & 31;
    const int half = lane >> 4;
    const int nloc = lane & 15;
    const int koff = half * 2;              // K sub-offset for 16x4 f32 frags
    const int mt   = blockIdx.x * 2 + wave; // row tile 0..47
    const int m0   = mt * 16;
    const float invT = tempv[0];
    const float rs   = rsl[layer];
    float* wlog = logits + (size_t)m0 * kT; // this tile's 16x768 logit rows

    // ---- preload A fragments of S rows m0..m0+15 (ISA 16x4 f32 layout) ----
    v2f aS[16];
    {
        const float* srow = S + (m0 + nloc) * kD;
        #pragma unroll
        for (int kk = 0; kk < 16; ++kk) {
            aS[kk].x = srow[4 * kk + koff];
            aS[kk].y = srow[4 * kk + koff + 1];
        }
    }

    const v8f zero8 = {0.f, 0.f, 0.f, 0.f, 0.f, 0.f, 0.f, 0.f};

    float rmax[8];
    #pragma unroll
    for (int r = 0; r < 8; ++r) rmax[r] = -3.0e38f;

    // ================= pass 1: logits + row max =================
    for (int j = 0; j < kNT; ++j) {
        __syncthreads();
        {   // stage S rows j*16..j*16+15 (contiguous 1024 floats)
            const float4* src = (const float4*)(S + j * 16 * kD);
            float4* dst = (float4*)tileBuf;
            for (int i = tid; i < 256; i += 64) dst[i] = src[i];
        }
        __syncthreads();

        v8f acc = zero8;
        #pragma unroll
        for (int kk = 0; kk < 16; ++kk) {
            v2f b;  // B[k][n] = S[j*16+n][k]
            b.x = tileBuf[nloc * kD + 4 * kk + koff];
            b.y = tileBuf[nloc * kD + 4 * kk + koff + 1];
            acc = __builtin_amdgcn_wmma_f32_16x16x4_f32(false, aS[kk], false, b,
                                                        (short)0, acc, false, false);
        }
        #pragma unroll
        for (int r = 0; r < 8; ++r) {
            float v = acc[r];
            v = fminf(fmaxf(v, -1.0f + 1e-6f), 1.0f - 1e-6f);
            float lg = -2.0f * acosf(v) * invT;
            wlog[(r + half * 8) * kT + j * 16 + nloc] = lg;
            rmax[r] = fmaxf(rmax[r], lg);
        }
    }

    // row max across the 16 lanes of each wave32 half
    #pragma unroll
    for (int r = 0; r < 8; ++r) {
        #pragma unroll
        for (int m = 1; m < 16; m <<= 1)
            rmax[r] = fmaxf(rmax[r], __shfl_xor(rmax[r], m, 32));
    }

    // ================= pass 2: exp/sum fused with O = P @ X =================
    float rsum[8];
    v8f O[4];
    #pragma unroll
    for (int r = 0; r < 8; ++r) rsum[r] = 0.0f;
    #pragma unroll
    for (int nt = 0; nt < 4; ++nt) O[nt] = zero8;

    float* myP = &Pt[wave][0];
    for (int j = 0; j < kNT; ++j) {
        __syncthreads();
        {   // stage X rows j*16..j*16+15
            const float4* src = (const float4*)(x + j * 16 * kD);
            float4* dst = (float4*)tileBuf;
            for (int i = tid; i < 256; i += 64) dst[i] = src[i];
        }
        // P tile: C-layout -> LDS row-major (padded stride 17)
        #pragma unroll
        for (int r = 0; r < 8; ++r) {
            float lg = wlog[(r + half * 8) * kT + j * 16 + nloc];
            float pv = __expf(lg - rmax[r]);
            rsum[r] += pv;
            myP[(r + half * 8) * 17 + nloc] = pv;
        }
        __syncthreads();

        #pragma unroll
        for (int kk = 0; kk < 4; ++kk) {
            v2f a;  // A[m][k] = P[m][4kk+koff..+1]
            a.x = myP[nloc * 17 + 4 * kk + koff];
            a.y = myP[nloc * 17 + 4 * kk + koff + 1];
            #pragma unroll
            for (int nt = 0; nt < 4; ++nt) {
                v2f b;  // B[k][n] = X[j*16+k][nt*16+n]
                b.x = tileBuf[(4 * kk + koff) * kD + nt * 16 + nloc];
                b.y = tileBuf[(4 * kk + koff + 1) * kD + nt * 16 + nloc];
                O[nt] = __builtin_amdgcn_wmma_f32_16x16x4_f32(false, a, false, b,
                                                              (short)0, O[nt], false, false);
            }
        }
    }

    // row sums across the 16 lanes of each wave32 half
    #pragma unroll
    for (int r = 0; r < 8; ++r) {
        #pragma unroll
        for (int m = 1; m < 16; m <<= 1)
            rsum[r] += __shfl_xor(rsum[r], m, 32);
    }

    // epilogue: x + rs*(attn - x)
    #pragma unroll
    for (int nt = 0; nt < 4; ++nt) {
        #pragma unroll
        for (int r = 0; r < 8; ++r) {
            int row = m0 + r + half * 8;
            int col = nt * 16 + nloc;
            float attn = O[nt][r] / rsum[r];
            float xin = x[row * kD + col];
            xout[row * kD + col] = xin + rs * (attn - xin);
        }
    }
}

// ---------------------------------------------------------------------------
// Basin update (after pass 0): pooled -> compress MLP -> gated basin update.
// Single block of 64 threads.
// ---------------------------------------------------------------------------
__global__ void basin_update_kernel(const float* __restrict__ x, const float* __restrict__ basin,
                                    const float* __restrict__ w1, const float* __restrict__ b1,
                                    const float* __restrict__ w2, const float* __restrict__ b2,
                                    const float* __restrict__ uw, const float* __restrict__ ub,
                                    float* __restrict__ basin_out) {
    __shared__ float pooled[kD];
    __shared__ float h1[kD / 2];
    __shared__ float agg[kD];
    const int d = threadIdx.x;

    float s = 0.0f;
    for (int t = 0; t < kT; ++t) s += x[t * kD + d];
    pooled[d] = s / (float)kT;
    __syncthreads();

    if (d < kD / 2) {
        float a = b1[d];
        #pragma unroll 8
        for (int k = 0; k < kD; ++k) a += w1[d * kD + k] * pooled[k];
        h1[d] = tanhf(a);
    }
    __syncthreads();

    {
        float a = b2[d];
        #pragma unroll 8
        for (int k = 0; k < kD / 2; ++k) a += w2[d * (kD / 2) + k] * h1[k];
        agg[d] = tanhf(a);
    }
    __syncthreads();

    float a = ub[d];
    const float* wr = uw + d * (2 * kD);
    #pragma unroll 8
    for (int k = 0; k < kD; ++k) a += basin[k] * wr[k] + agg[k] * wr[kD + k];
    float g = sigmoidf_(a);
    basin_out[d] = basin[d] * (1.0f - g) + agg[d] * g;
}

// ---------------------------------------------------------------------------
// out = x + 0.01 * res_scale * (x - basin_seq)
// ---------------------------------------------------------------------------
__global__ void final_kernel(const float* __restrict__ x, const float* __restrict__ bs,
                             const float* __restrict__ rsc, float* __restrict__ out) {
    const int i = blockIdx.x * kD + threadIdx.x;
    float xi = x[i];
    out[i] = xi + 0.01f * rsc[0] * (xi - bs[i]);
}

// ---------------------------------------------------------------------------
// Host orchestration (graph-capture safe: launches + D2D async copies only)
// ---------------------------------------------------------------------------
extern "C" void kernel_launch(void* const* d_in, const int* in_sizes, int n_in,
                              void* d_out, int out_size, void* d_ws, size_t ws_size,
                              hipStream_t stream) {
    (void)in_sizes; (void)n_in; (void)out_size; (void)ws_size;

    const float* basin_seq    = (const float*)d_in[0];   // (T,D)
    const float* basin_coords = (const float*)d_in[1];   // (D,)
    const float* temp_w       = (const float*)d_in[2];   // (L,D)
    const float* temp_b       = (const float*)d_in[3];   // (L,)
    const float* rsl          = (const float*)d_in[4];   // (L,)
    const float* fb_w         = (const float*)d_in[5];   // (L,D,2D)
    const float* fb_b         = (const float*)d_in[6];   // (L,D)
    const float* cw1          = (const float*)d_in[7];   // (D/2,D)
    const float* cb1          = (const float*)d_in[8];   // (D/2,)
    const float* cw2          = (const float*)d_in[9];   // (D,D/2)
    const float* cb2          = (const float*)d_in[10];  // (D,)
    const float* uw           = (const float*)d_in[11];  // (D,2D)
    const float* ub           = (const float*)d_in[12];  // (D,)
    const float* res_scale    = (const float*)d_in[13];  // scalar

    constexpr size_t NX = (size_t)kT * kD;               // 49152 floats
    float* W       = (float*)d_ws;
    float* xA      = W;
    float* xB      = W + NX;
    float* Sb      = W + 2 * NX;
    float* prevs   = W + 3 * NX;                         // 4 x NX
    float* logitsW = W + 7 * NX;                         // T*T floats
    float* tempv   = W + 7 * NX + (size_t)kT * kT;
    float* basinB  = tempv + 4;

    float* xcur = xA;
    float* xalt = xB;
    hipMemcpyAsync(xcur, basin_seq, NX * sizeof(float), hipMemcpyDeviceToDevice, stream);

    const float* basin_cur = basin_coords;

    constexpr int PASSES = 2;
    for (int p = 0; p < PASSES; ++p) {
        for (int l = 0; l < kL; ++l) {
            if (p > 0) {
                fb_gate_kernel<<<kT, kD, 0, stream>>>(xcur, prevs + (size_t)l * NX,
                                                      fb_w + (size_t)l * kD * 2 * kD,
                                                      fb_b + (size_t)l * kD, xalt);
                float* t = xcur; xcur = xalt; xalt = t;
            }
            temp_kernel<<<1, kD, 0, stream>>>(basin_cur, temp_w + (size_t)l * kD,
                                              temp_b + l, tempv);
            simplex_sqrt_kernel<<<kT, kD, 0, stream>>>(xcur, Sb);
            attn_kernel<<<kNT / 2, 64, 0, stream>>>(xcur, Sb, logitsW, tempv, rsl, l, xalt);
            { float* t = xcur; xcur = xalt; xalt = t; }
            if (p == 0) {
                hipMemcpyAsync(prevs + (size_t)l * NX, xcur, NX * sizeof(float),
                               hipMemcpyDeviceToDevice, stream);
            }
        }
        if (p == 0) {
            basin_update_kernel<<<1, kD, 0, stream>>>(xcur, basin_cur, cw1, cb1,
                                                      cw2, cb2, uw, ub, basinB);
            basin_cur = basinB;
        }
    }

    final_kernel<<<kT, kD, 0, stream>>>(xcur, basin_seq, res_scale, (float*)d_out);
}